// MambaBlock_25786983646113
// MI455X (gfx1250) — compile-verified
//
#include <hip/hip_runtime.h>
#include <hip/hip_bf16.h>

// Mamba block for MI455X (gfx1250), fp32 throughout using V_WMMA_F32_16X16X4_F32
// for all GEMMs (fp32 precision needed: output goes through LayerNorm).
// Pipeline: gemm_in -> conv+silu -> gemm_xdbl -> gemm_delta(+softplus)
//           -> selective scan (1 wave32 per (b,channel), 2 states/lane)
//           -> gemm_out(+residual) -> layernorm.

#define D_MODEL 768
#define D_STATE 64
#define D_CONV  4
#define D_INNER 1536
#define DT_RANK 48
#define XDBL_N  176          // DT_RANK + 2*D_STATE
#define BATCH   2
#define SEQLEN  4096
#define NTOK    (BATCH * SEQLEN)   // 8192
#define LN_EPS  1e-5f

typedef float v2f __attribute__((ext_vector_type(2)));
typedef float v8f __attribute__((ext_vector_type(8)));

__device__ __forceinline__ v8f wmma_f32(v2f a, v2f b, v8f c) {
  // D = A(16x4 f32) * B(4x16 f32) + C(16x16 f32)
  return __builtin_amdgcn_wmma_f32_16x16x4_f32(false, a, false, b, (short)0, c,
                                               false, false);
}

// One wave computes a 16x(NT*16) strip of C = A * Bw^T.
// A: row-major [M, lda], Bw: row-major [N, ldb] (weights), K = reduction dim.
// Fragment layout (wave32): lane l -> row m = l%16, k-half = 2*(l/16),
// VGPR j holds k = khalf + j.  Identical addressing for A rows and Bw rows.
template <int NT>
__device__ __forceinline__ void gemm_tile(const float* __restrict__ A, int lda,
                                          const float* __restrict__ Bw, int ldb,
                                          int K, int m0, int n0, v8f (&acc)[NT]) {
  const int lane = threadIdx.x & 31;
  const int r    = lane & 15;
  const int kh   = (lane >> 4) << 1;   // 0 or 2
  const float* ap = A  + (size_t)(m0 + r) * lda + kh;
  const float* bp = Bw + (size_t)(n0 + r) * ldb + kh;
  const v8f vzero = {0.f, 0.f, 0.f, 0.f, 0.f, 0.f, 0.f, 0.f};
#pragma unroll
  for (int nt = 0; nt < NT; ++nt) acc[nt] = vzero;
#pragma unroll 4
  for (int k = 0; k < K; k += 4) {
    v2f a = *(const v2f*)(ap + k);
#pragma unroll
    for (int nt = 0; nt < NT; ++nt) {
      v2f b = *(const v2f*)(bp + (size_t)(nt * 16) * ldb + k);
      acc[nt] = wmma_f32(a, b, acc[nt]);
    }
  }
}

// ---------------- Kernel 1: xz = x @ W_in^T, split into xi / z ----------------
__global__ __launch_bounds__(256) void gemm_in_kernel(
    const float* __restrict__ X, const float* __restrict__ W_in,
    float* __restrict__ xi, float* __restrict__ z) {
  const int wave = (int)((blockIdx.x * blockDim.x + threadIdx.x) >> 5);
  const int NGRP = (2 * D_INNER) / 64;          // 48 groups of 4 n-tiles
  const int tm = wave / NGRP;
  const int tg = wave % NGRP;
  if (tm >= NTOK / 16) return;
  const int m0 = tm * 16, n0 = tg * 64;
  v8f acc[4];
  gemm_tile<4>(X, D_MODEL, W_in, D_MODEL, D_MODEL, m0, n0, acc);
  const int lane = threadIdx.x & 31;
  const int col  = lane & 15;
  const int row0 = m0 + ((lane >> 4) << 3);
#pragma unroll
  for (int nt = 0; nt < 4; ++nt) {
    const int gcol = n0 + nt * 16 + col;
    float* dst = (gcol < D_INNER) ? xi : z;
    const int c = (gcol < D_INNER) ? gcol : (gcol - D_INNER);
#pragma unroll
    for (int j = 0; j < 8; ++j)
      dst[(size_t)(row0 + j) * D_INNER + c] = acc[nt][j];
  }
}

// ---------------- Kernel 2: depthwise causal conv + bias + SiLU --------------
__global__ __launch_bounds__(256) void conv_silu_kernel(
    const float* __restrict__ xi, const float* __restrict__ conv_w,
    const float* __restrict__ conv_b, float* __restrict__ u) {
  const size_t idx = (size_t)blockIdx.x * blockDim.x + threadIdx.x;
  const size_t total = (size_t)NTOK * D_INNER;
  if (idx >= total) return;
  const int d = (int)(idx % D_INNER);
  const size_t bt = idx / D_INNER;
  const int t = (int)(bt % SEQLEN);
  float acc = conv_b[d];
#pragma unroll
  for (int j = 0; j < D_CONV; ++j) {
    const int tt = t - (D_CONV - 1) + j;
    if (tt >= 0)
      acc += conv_w[d * D_CONV + j] *
             xi[(bt - (size_t)(D_CONV - 1) + j) * D_INNER + d];
  }
  u[idx] = acc / (1.f + expf(-acc));   // silu
}

// ---------------- Kernel 3: x_dbl = u @ W_x^T  (N=176, NT=1) -----------------
__global__ __launch_bounds__(256) void gemm_xdbl_kernel(
    const float* __restrict__ u, const float* __restrict__ W_x,
    float* __restrict__ xdbl) {
  const int wave = (int)((blockIdx.x * blockDim.x + threadIdx.x) >> 5);
  const int NTILES = XDBL_N / 16;               // 11
  const int tm = wave / NTILES;
  const int tn = wave % NTILES;
  if (tm >= NTOK / 16) return;
  v8f acc[1];
  gemm_tile<1>(u, D_INNER, W_x, D_INNER, D_INNER, tm * 16, tn * 16, acc);
  const int lane = threadIdx.x & 31;
  const int gcol = tn * 16 + (lane & 15);
  const int row0 = tm * 16 + ((lane >> 4) << 3);
#pragma unroll
  for (int j = 0; j < 8; ++j)
    xdbl[(size_t)(row0 + j) * XDBL_N + gcol] = acc[0][j];
}

// ------- Kernel 4: delta = softplus(dt @ W_dt^T + b_dt)  (K=48, N=1536) ------
__global__ __launch_bounds__(256) void gemm_delta_kernel(
    const float* __restrict__ xdbl, const float* __restrict__ W_dt,
    const float* __restrict__ b_dt, float* __restrict__ delta) {
  const int wave = (int)((blockIdx.x * blockDim.x + threadIdx.x) >> 5);
  const int NGRP = D_INNER / 64;                // 24
  const int tm = wave / NGRP;
  const int tg = wave % NGRP;
  if (tm >= NTOK / 16) return;
  const int m0 = tm * 16, n0 = tg * 64;
  v8f acc[4];
  // dt occupies columns [0, 48) of x_dbl rows (lda = 176)
  gemm_tile<4>(xdbl, XDBL_N, W_dt, DT_RANK, DT_RANK, m0, n0, acc);
  const int lane = threadIdx.x & 31;
  const int col  = lane & 15;
  const int row0 = m0 + ((lane >> 4) << 3);
#pragma unroll
  for (int nt = 0; nt < 4; ++nt) {
    const int gcol = n0 + nt * 16 + col;
    const float bb = b_dt[gcol];
#pragma unroll
    for (int j = 0; j < 8; ++j) {
      const float v = acc[nt][j] + bb;
      delta[(size_t)(row0 + j) * D_INNER + gcol] =
          (v > 20.f) ? v : log1pf(expf(v));   // stable softplus
    }
  }
}

// ---------------- Kernel 5: selective scan + Dp skip + silu(z) gate ----------
// One wave32 per (b, d). Lane owns states n=lane and n=lane+32.
__global__ __launch_bounds__(256) void scan_kernel(
    const float* __restrict__ delta, const float* __restrict__ u,
    const float* __restrict__ xdbl, const float* __restrict__ z,
    const float* __restrict__ A_log, const float* __restrict__ Dp,
    float* __restrict__ yact) {
  const int wave = (int)((blockIdx.x * blockDim.x + threadIdx.x) >> 5);
  const int lane = threadIdx.x & 31;
  if (wave >= BATCH * D_INNER) return;
  const int b = wave / D_INNER;
  const int d = wave % D_INNER;
  const int n0 = lane, n1 = lane + 32;
  const float nA0 = -expf(A_log[(size_t)d * D_STATE + n0]);  // A = -exp(A_log)
  const float nA1 = -expf(A_log[(size_t)d * D_STATE + n1]);
  const float Dd = Dp[d];
  float h0 = 0.f, h1 = 0.f;
  const size_t rowB = (size_t)b * SEQLEN;
  for (int t = 0; t < SEQLEN; ++t) {
    const size_t tok = rowB + t;
    const float* xr = xdbl + tok * XDBL_N;
    const float dt_ = delta[tok * D_INNER + d];
    const float u_  = u[tok * D_INNER + d];
    const float B0 = xr[DT_RANK + n0];
    const float B1 = xr[DT_RANK + n1];
    const float C0 = xr[DT_RANK + D_STATE + n0];
    const float C1 = xr[DT_RANK + D_STATE + n1];
    if (t + 1 < SEQLEN)                          // global_prefetch_b8
      __builtin_prefetch(xdbl + (tok + 1) * XDBL_N + DT_RANK + lane, 0, 0);
    const float du = dt_ * u_;
    h0 = h0 * __expf(dt_ * nA0) + du * B0;
    h1 = h1 * __expf(dt_ * nA1) + du * B1;
    float p = h0 * C0 + h1 * C1;
#pragma unroll
    for (int off = 16; off > 0; off >>= 1) p += __shfl_xor(p, off, 32);
    if (lane == 0) {
      const float zv = z[tok * D_INNER + d];
      const float sz = zv / (1.f + expf(-zv));   // silu(z)
      yact[tok * D_INNER + d] = (p + u_ * Dd) * sz;
    }
  }
}

// ---------------- Kernel 6: r = x + yact @ W_out^T  (N=768) ------------------
__global__ __launch_bounds__(256) void gemm_out_kernel(
    const float* __restrict__ yact, const float* __restrict__ W_out,
    const float* __restrict__ X, float* __restrict__ r) {
  const int wave = (int)((blockIdx.x * blockDim.x + threadIdx.x) >> 5);
  const int NGRP = D_MODEL / 64;                // 12
  const int tm = wave / NGRP;
  const int tg = wave % NGRP;
  if (tm >= NTOK / 16) return;
  const int m0 = tm * 16, n0 = tg * 64;
  v8f acc[4];
  gemm_tile<4>(yact, D_INNER, W_out, D_INNER, D_INNER, m0, n0, acc);
  const int lane = threadIdx.x & 31;
  const int col  = lane & 15;
  const int row0 = m0 + ((lane >> 4) << 3);
#pragma unroll
  for (int nt = 0; nt < 4; ++nt) {
    const int gcol = n0 + nt * 16 + col;
#pragma unroll
    for (int j = 0; j < 8; ++j) {
      const size_t o = (size_t)(row0 + j) * D_MODEL + gcol;
      r[o] = acc[nt][j] + X[o];                  // residual
    }
  }
}

// ---------------- Kernel 7: LayerNorm over last dim (768) --------------------
__global__ __launch_bounds__(256) void layernorm_kernel(
    const float* __restrict__ r, const float* __restrict__ ln_w,
    const float* __restrict__ ln_b, float* __restrict__ out) {
  __shared__ float s_sum[8];
  __shared__ float s_sq[8];
  const int row = blockIdx.x;
  const float* rr = r + (size_t)row * D_MODEL;
  float v[3];
  float sum = 0.f, sq = 0.f;
#pragma unroll
  for (int i = 0; i < 3; ++i) {
    v[i] = rr[threadIdx.x + i * 256];
    sum += v[i];
    sq  += v[i] * v[i];
  }
#pragma unroll
  for (int off = 16; off > 0; off >>= 1) {
    sum += __shfl_xor(sum, off, 32);
    sq  += __shfl_xor(sq,  off, 32);
  }
  const int wid = threadIdx.x >> 5, lane = threadIdx.x & 31;
  if (lane == 0) { s_sum[wid] = sum; s_sq[wid] = sq; }
  __syncthreads();
  sum = 0.f; sq = 0.f;
#pragma unroll
  for (int i = 0; i < 8; ++i) { sum += s_sum[i]; sq += s_sq[i]; }
  const float mu  = sum * (1.f / D_MODEL);
  const float var = sq * (1.f / D_MODEL) - mu * mu;
  const float inv = rsqrtf(var + LN_EPS);
#pragma unroll
  for (int i = 0; i < 3; ++i) {
    const int c = threadIdx.x + i * 256;
    out[(size_t)row * D_MODEL + c] = (v[i] - mu) * inv * ln_w[c] + ln_b[c];
  }
}

extern "C" void kernel_launch(void* const* d_in, const int* in_sizes, int n_in,
                              void* d_out, int out_size, void* d_ws,
                              size_t ws_size, hipStream_t stream) {
  (void)in_sizes; (void)n_in; (void)out_size; (void)ws_size;
  const float* x      = (const float*)d_in[0];
  const float* W_in   = (const float*)d_in[1];
  const float* conv_w = (const float*)d_in[2];
  const float* conv_b = (const float*)d_in[3];
  const float* W_x    = (const float*)d_in[4];
  const float* W_dt   = (const float*)d_in[5];
  const float* b_dt   = (const float*)d_in[6];
  const float* A_log  = (const float*)d_in[7];
  const float* Dp     = (const float*)d_in[8];
  const float* W_out  = (const float*)d_in[9];
  const float* ln_w   = (const float*)d_in[10];
  const float* ln_b   = (const float*)d_in[11];
  float* out = (float*)d_out;

  float* ws = (float*)d_ws;
  size_t off = 0;
  float* xi    = ws + off; off += (size_t)NTOK * D_INNER;
  float* z     = ws + off; off += (size_t)NTOK * D_INNER;
  float* u     = ws + off; off += (size_t)NTOK * D_INNER;
  float* delta = ws + off; off += (size_t)NTOK * D_INNER;
  float* xdbl  = ws + off; off += (size_t)NTOK * XDBL_N;
  float* yact  = xi;   // xi dead after conv
  float* rbuf  = z;    // z dead after scan

  gemm_in_kernel<<<(NTOK / 16) * 48 / 8, 256, 0, stream>>>(x, W_in, xi, z);
  conv_silu_kernel<<<(NTOK * D_INNER) / 256, 256, 0, stream>>>(xi, conv_w,
                                                               conv_b, u);
  gemm_xdbl_kernel<<<(NTOK / 16) * 11 / 8, 256, 0, stream>>>(u, W_x, xdbl);
  gemm_delta_kernel<<<(NTOK / 16) * 24 / 8, 256, 0, stream>>>(xdbl, W_dt, b_dt,
                                                              delta);
  scan_kernel<<<(BATCH * D_INNER) / 8, 256, 0, stream>>>(delta, u, xdbl, z,
                                                         A_log, Dp, yact);
  gemm_out_kernel<<<(NTOK / 16) * 12 / 8, 256, 0, stream>>>(yact, W_out, x,
                                                            rbuf);
  layernorm_kernel<<<NTOK, 256, 0, stream>>>(rbuf, ln_w, ln_b, out);
}